// FTT_Solver_60473139528511
// MI455X (gfx1250) — compile-verified
//
#include <hip/hip_runtime.h>
#include <hip/hip_bf16.h>

typedef float v2f __attribute__((ext_vector_type(2)));
typedef float v8f __attribute__((ext_vector_type(8)));
typedef unsigned int v4u __attribute__((ext_vector_type(4)));
typedef int v4i __attribute__((ext_vector_type(4)));
typedef int v8i __attribute__((ext_vector_type(8)));

#define NSWEEPS 8

__device__ __forceinline__ v8f wmma4(v2f a, v2f b, v8f c) {
  // D = A(16x4,f32) * B(4x16,f32) + C(16x16,f32)
  return __builtin_amdgcn_wmma_f32_16x16x4_f32(false, a, false, b, (short)0, c,
                                               false, false);
}

// A-fragment from a row-major [16 x ld] matrix: A[m][k0..k0+3]
__device__ __forceinline__ v2f frag_a(const float* M, int ld, int k0, int lane) {
  int m = lane & 15;
  int kb = k0 + ((lane >> 4) << 1);
  v2f r;
  r.x = M[m * ld + kb];
  r.y = M[m * ld + kb + 1];
  return r;
}

// B-fragment from a row-major [K x ld] matrix: B[k0..k0+3][n0..n0+15]
__device__ __forceinline__ v2f frag_b(const float* M, int ld, int k0, int n0, int lane) {
  int n = n0 + (lane & 15);
  int kb = k0 + ((lane >> 4) << 1);
  v2f r;
  r.x = M[kb * ld + n];
  r.y = M[(kb + 1) * ld + n];
  return r;
}

// Parallel cyclic Jacobi eigendecomposition of symmetric 16x16 sA.
// Accumulates eigenvectors (columns) into sV. Block-wide (256 threads).
__device__ void jacobi16(float* sA, float* sV, float* sPc, float* sPs,
                         int* sPp, int* sPq, int tid) {
  sV[tid] = ((tid >> 4) == (tid & 15)) ? 1.0f : 0.0f;
  __syncthreads();
  for (int sweep = 0; sweep < NSWEEPS; ++sweep) {
    for (int round = 0; round < 15; ++round) {
      if (tid < 8) {
        int p, q;
        if (tid == 0) {
          p = 0;
          q = 1 + round;
        } else {
          p = 1 + ((round + tid) % 15);
          q = 1 + ((round - tid + 15) % 15);
        }
        if (p > q) { int t = p; p = q; q = t; }
        float app = sA[p * 16 + p], aqq = sA[q * 16 + q], apq = sA[p * 16 + q];
        float c = 1.0f, s = 0.0f;
        if (fabsf(apq) > 1e-20f) {
          float tau = (aqq - app) / (2.0f * apq);
          float t = (tau >= 0.0f ? 1.0f : -1.0f) /
                    (fabsf(tau) + sqrtf(1.0f + tau * tau));
          c = rsqrtf(1.0f + t * t);
          s = t * c;
        }
        sPp[tid] = p; sPq[tid] = q; sPc[tid] = c; sPs[tid] = s;
      }
      __syncthreads();
      if (tid < 128) {  // row rotations: A <- J^T A  (disjoint rows per pair)
        int pi = tid >> 4, col = tid & 15;
        int p = sPp[pi], q = sPq[pi];
        float c = sPc[pi], s = sPs[pi];
        float ap = sA[p * 16 + col], aq = sA[q * 16 + col];
        sA[p * 16 + col] = c * ap - s * aq;
        sA[q * 16 + col] = s * ap + c * aq;
      }
      __syncthreads();
      if (tid < 128) {  // col rotations: A <- A J, V <- V J
        int pi = tid >> 4, row = tid & 15;
        int p = sPp[pi], q = sPq[pi];
        float c = sPc[pi], s = sPs[pi];
        float ap = sA[row * 16 + p], aq = sA[row * 16 + q];
        sA[row * 16 + p] = c * ap - s * aq;
        sA[row * 16 + q] = s * ap + c * aq;
        float vp = sV[row * 16 + p], vq = sV[row * 16 + q];
        sV[row * 16 + p] = c * vp - s * vq;
        sV[row * 16 + q] = s * vp + c * vq;
      }
      __syncthreads();
    }
  }
}

__device__ void eig_sort(const float* sA, float* sLam, int* sPerm, int tid) {
  if (tid < 16) sLam[tid] = sA[tid * 16 + tid];
  __syncthreads();
  if (tid < 16) {
    float v = sLam[tid];
    int r = 0;
    for (int j = 0; j < 16; ++j) {
      float u = sLam[j];
      if (u > v || (u == v && j < tid)) r++;
    }
    sPerm[r] = tid;
  }
  __syncthreads();
}

// b_full[i*256+j*16+k] = sum_{a,bb} b0[i,a] b1[a,j,bb] b2[bb,k]
__global__ void __launch_bounds__(256)
bias_full_kernel(const float* __restrict__ b0, const float* __restrict__ b1,
                 const float* __restrict__ b2, float* __restrict__ bfull) {
  int tid = threadIdx.x;
  for (int o = tid; o < 4096; o += 256) {
    int i = o >> 8, j = (o >> 4) & 15, k = o & 15;
    float acc = 0.0f;
#pragma unroll
    for (int a = 0; a < 2; ++a)
#pragma unroll
      for (int bb = 0; bb < 2; ++bb)
        acc += b0[i * 2 + a] * b1[a * 32 + j * 2 + bb] * b2[bb * 16 + k];
    bfull[o] = acc;
  }
}

__global__ void __launch_bounds__(256)
ftt_solver_kernel(const float* __restrict__ x0, const float* __restrict__ x1,
                  const float* __restrict__ x2, const float* __restrict__ w0,
                  const float* __restrict__ w1, const float* __restrict__ w2,
                  const float* __restrict__ bfull, float* __restrict__ out,
                  int B) {
  __shared__ float sX0[128], sX1[1024], sX2[128];
  __shared__ float sY0[1024];   // [p][a*8+e]            16x64
  __shared__ float sY2[1024];   // [(a*8+e)][k]          64x16
  __shared__ float sU[2048];    // u_c [16][128] (also tx scratch)
  __shared__ float sT[2048];    // t_c [16][128]
  __shared__ float sZ[4096];    // z   [16][256]
  __shared__ float sRest[2048]; // rest [128][16]
  __shared__ float sA[256], sV[256], sU1[256];
  __shared__ float sS1[16], sS2[16], sLam[16];
  __shared__ int sPerm[16];
  __shared__ float sPc[8], sPs[8];
  __shared__ int sPp[8], sPq[8];
  __shared__ float sF;

  const int tid = threadIdx.x;
  const int lane = tid & 31;
  const int wv = tid >> 5;  // 8 waves
  const long b = blockIdx.x;

  const float L = 1.0f, Sconst = -1.0f, EPS = 1.0f;

  // ---- phase 0: stage per-sample x cores.
  // x1 tile (4KB contiguous) is staged into LDS by the Tensor Data Mover
  // (TDM) from wave 0 while the rest of the block copies x0/x2; the TDM
  // transfer is tracked with TENSORcnt (s_wait_tensorcnt) per ISA ch.7/8.
#if __has_builtin(__builtin_amdgcn_tensor_load_to_lds)
  if (wv == 0) {
    unsigned lds_byte = (unsigned)(uintptr_t)(void*)&sX1[0];  // LDS aperture: addr[31:0]
    unsigned long long ga = (unsigned long long)(uintptr_t)(x1 + b * 1024);
    v4u g0;
    g0.x = 1u;                                   // count=1, is_restore=0, gather=0
    g0.y = lds_byte;                             // lds_addr
    g0.z = (unsigned)(ga & 0xffffffffu);         // global_addr[31:0]
    g0.w = (unsigned)((ga >> 32) & 0x1ffffffu)   // global_addr[56:32]
           | (2u << 30);                         // type=2 ("image")
    v8i g1;
    g1[0] = (2 << 16);           // workgroup_mask=0, data_size=2 (4 bytes)
    g1[1] = (int)((1024u & 0xffffu) << 16);  // tensor_dim0[15:0] @ bit48
    g1[2] = (int)(1u << 16);     // tensor_dim0[31:16]=0 | tensor_dim1=1
    g1[3] = (int)(1024u << 16);  // tensor_dim1 hi=0 | tile_dim0=1024
    g1[4] = 1;                   // tile_dim1=1, tile_dim2=0
    g1[5] = 1024;                // tensor_dim0_stride[31:0]
    g1[6] = 0;                   // stride hi | tensor_dim1_stride lo
    g1[7] = 0;
    v4i g2 = {0, 0, 0, 0};       // group2/3 unused (<=2D tensor)
    v4i g3 = {0, 0, 0, 0};
#if defined(__clang_major__) && (__clang_major__ >= 23)
    v8i g4 = {0, 0, 0, 0, 0, 0, 0, 0};
    __builtin_amdgcn_tensor_load_to_lds(g0, g1, g2, g3, g4, 0);
#else
    __builtin_amdgcn_tensor_load_to_lds(g0, g1, g2, g3, 0);
#endif
    __builtin_amdgcn_s_wait_tensorcnt(0);
  }
#else
  for (int i = tid; i < 1024; i += 256) sX1[i] = x1[b * 1024 + i];
#endif
  if (tid < 128) {
    sX0[tid] = x0[b * 128 + tid];
    sX2[tid] = x2[b * 128 + tid];
  }
  if (tid == 0) sF = 0.0f;
  __builtin_prefetch(w1 + tid * 64, 0, 1);  // warm L2/WGP$ for the B matrix
  __syncthreads();

  // ---- phase 1: f_norm = ||x_full||^2  (tx in sU scratch)
  for (int e = tid; e < 2048; e += 256) {
    int i = e >> 7, rem = e & 127;  // rem = j*8+a2
    float acc = 0.0f;
#pragma unroll
    for (int a = 0; a < 8; ++a) acc += sX0[i * 8 + a] * sX1[a * 128 + rem];
    sU[e] = acc;
  }
  __syncthreads();
  {
    float part = 0.0f;
    for (int e = tid; e < 4096; e += 256) {
      int k = e & 15, ij = e >> 4;
      float acc = 0.0f;
#pragma unroll
      for (int a2 = 0; a2 < 8; ++a2) acc += sU[ij * 8 + a2] * sX2[a2 * 16 + k];
      part += acc * acc;
    }
    atomicAdd(&sF, part);
  }
  __syncthreads();
  const float s_val = (sF > EPS) ? 0.0f : Sconst;
  __syncthreads();

  // ---- phase 2: boundary cores y0 [16x64], y2 [64x16]
  for (int o = tid; o < 1024; o += 256) {
    int p = o >> 6, ae = o & 63, a = ae >> 3, e = ae & 7;
    float acc = 0.0f;
#pragma unroll
    for (int nn = 0; nn < 16; ++nn)
      acc += sX0[nn * 8 + a] * w0[nn * 128 + p * 8 + e];
    sY0[p * 64 + ae] = acc;
  }
  for (int o = tid; o < 1024; o += 256) {
    int ae = o >> 4, k = o & 15, a = ae >> 3, e = ae & 7;
    float acc = 0.0f;
#pragma unroll
    for (int nn = 0; nn < 16; ++nn)
      acc += sX2[a * 16 + nn] * w2[e * 256 + nn * 16 + k];
    sY2[ae * 16 + k] = acc;
  }

  // z accumulators: two 16x16 C tiles per wave (rows i=wv and i=wv+8 of z[(i,j),k])
  v8f zacc0, zacc1;
  {
    int h = lane >> 4, col = lane & 15;
#pragma unroll
    for (int r = 0; r < 8; ++r) {
      int mg0 = wv * 16 + r + 8 * h;
      int mg1 = (wv + 8) * 16 + r + 8 * h;
      zacc0[r] = bfull[mg0 * 16 + col] * (1.0f / L) + s_val;
      zacc1[r] = bfull[mg1 * 16 + col] * (1.0f / L) + s_val;
    }
  }
  __syncthreads();

  // ---- phase 3: rank contraction, tiled over c (x-rank of the right TT bond)
  for (int c = 0; c < 8; ++c) {
    // u_c[i][e*16+n] = sum_a y0[i][a*8+e] * x1[a][n][c]
    for (int o = tid; o < 2048; o += 256) {
      int i = o >> 7, en = o & 127, e = en >> 4, nn = en & 15;
      float acc = 0.0f;
#pragma unroll
      for (int a = 0; a < 8; ++a)
        acc += sY0[i * 64 + a * 8 + e] * sX1[a * 128 + nn * 8 + c];
      sU[i * 128 + en] = acc;
    }
    __syncthreads();

    // t_c = u_c(16x128) * w1(128x128): wave wv owns 16-column tile
    {
      v8f tacc = {0.f, 0.f, 0.f, 0.f, 0.f, 0.f, 0.f, 0.f};
      int n0 = wv * 16;
      for (int ks = 0; ks < 32; ++ks) {
        v2f af = frag_a(sU, 128, ks * 4, lane);
        v2f bf = frag_b(w1, 128, ks * 4, n0, lane);
        tacc = wmma4(af, bf, tacc);
      }
      int h = lane >> 4, nn = n0 + (lane & 15);
#pragma unroll
      for (int r = 0; r < 8; ++r) sT[(r + 8 * h) * 128 + nn] = tacc[r];
    }
    __syncthreads();

    // z[(i,j)][k] += sum_f t_c[i][j*8+f] * y2[(c*8+f)][k]   (K=8 -> 2 steps)
#pragma unroll
    for (int half = 0; half < 2; ++half) {
      int k0 = half * 4;
      int kb = k0 + ((lane >> 4) << 1);
      int m = lane & 15;
      v2f bf;
      bf.x = sY2[(c * 8 + kb) * 16 + (lane & 15)];
      bf.y = sY2[(c * 8 + kb + 1) * 16 + (lane & 15)];
      v2f a0, a1;  // A gathered: row mg=(i,j) -> t_c[i*128 + j*8 + f]
      a0.x = sT[wv * 128 + m * 8 + kb];
      a0.y = sT[wv * 128 + m * 8 + kb + 1];
      a1.x = sT[(wv + 8) * 128 + m * 8 + kb];
      a1.y = sT[(wv + 8) * 128 + m * 8 + kb + 1];
      zacc0 = wmma4(a0, bf, zacc0);
      zacc1 = wmma4(a1, bf, zacc1);
    }
    __syncthreads();
  }

  // ---- phase 4: spill z tiles to LDS (z[i*256 + j*16 + k] == m1[i][col])
  {
    int h = lane >> 4, col = lane & 15;
#pragma unroll
    for (int r = 0; r < 8; ++r) {
      sZ[(wv * 16 + r + 8 * h) * 16 + col] = zacc0[r];
      sZ[((wv + 8) * 16 + r + 8 * h) * 16 + col] = zacc1[r];
    }
  }
  __syncthreads();

  // ---- phase 5: Gram1 = m1 m1^T via WMMA (A-frag == B-frag), then Jacobi
  if (wv == 0) {
    v8f g = {0.f, 0.f, 0.f, 0.f, 0.f, 0.f, 0.f, 0.f};
    for (int ks = 0; ks < 64; ++ks) {
      int kb = ks * 4 + ((lane >> 4) << 1);
      int m = lane & 15;
      v2f f;
      f.x = sZ[m * 256 + kb];
      f.y = sZ[m * 256 + kb + 1];
      g = wmma4(f, f, g);
    }
    int h = lane >> 4, colj = lane & 15;
#pragma unroll
    for (int r = 0; r < 8; ++r) sA[(r + 8 * h) * 16 + colj] = g[r];
  }
  __syncthreads();
  jacobi16(sA, sV, sPc, sPs, sPp, sPq, tid);
  eig_sort(sA, sLam, sPerm, tid);
  if (tid < 16) sS1[tid] = sqrtf(fmaxf(sLam[sPerm[tid]], 0.0f));
  __syncthreads();

  float* oU1 = out + (size_t)B * 1280 + b * 256;
  float* oc0 = out + b * 128;
  float* oS1 = out + (size_t)B * 1536 + b * 16;
  {
    int i = tid >> 4, r = tid & 15;
    float u = sV[i * 16 + sPerm[r]];
    sU1[i * 16 + r] = u;
    oU1[i * 16 + r] = u;
    if (r < 8) oc0[i * 8 + r] = u;  // c0r = U1[:, :8]
    if (tid < 16) oS1[tid] = sS1[tid];
  }
  __syncthreads();

  // ---- phase 6: V1 = (U1^T m1)/S1 ; rest = S1[:8]*V1[:8] = U1[:, :8]^T m1
  float* oV1 = out + (size_t)B * 1552 + b * 4096;
  for (int o = tid; o < 4096; o += 256) {
    int r = o >> 8, col = o & 255;
    float acc = 0.0f;
#pragma unroll
    for (int i = 0; i < 16; ++i) acc += sU1[i * 16 + r] * sZ[i * 256 + col];
    float sv = sS1[r];
    float rin = (sv > 1e-12f) ? (1.0f / sv) : 0.0f;
    oV1[r * 256 + col] = acc * rin;
    if (r < 8) sRest[r * 256 + col] = acc;  // rest[128][16] row-major
  }
  __syncthreads();

  // ---- phase 7: Gram2 = rest^T rest via WMMA, Jacobi, outputs
  if (wv == 0) {
    v8f g = {0.f, 0.f, 0.f, 0.f, 0.f, 0.f, 0.f, 0.f};
    for (int ks = 0; ks < 32; ++ks) {
      int kb = ks * 4 + ((lane >> 4) << 1);
      v2f f;
      f.x = sRest[kb * 16 + (lane & 15)];
      f.y = sRest[(kb + 1) * 16 + (lane & 15)];
      g = wmma4(f, f, g);
    }
    int h = lane >> 4, colj = lane & 15;
#pragma unroll
    for (int r = 0; r < 8; ++r) sA[(r + 8 * h) * 16 + colj] = g[r];
  }
  __syncthreads();
  jacobi16(sA, sV, sPc, sPs, sPp, sPq, tid);
  eig_sort(sA, sLam, sPerm, tid);

  float* oS2 = out + (size_t)B * 7696 + b * 16;
  float* oV2 = out + (size_t)B * 7712 + b * 256;
  float* oc2 = out + (size_t)B * 1152 + b * 128;
  if (tid < 16) {
    float sv = sqrtf(fmaxf(sLam[sPerm[tid]], 0.0f));
    sS2[tid] = sv;
    oS2[tid] = sv;
  }
  __syncthreads();
  {
    int r = tid >> 4, nn = tid & 15;
    float wval = sV[nn * 16 + sPerm[r]];  // V2 (Vh) row r = eigvec perm[r]
    oV2[r * 16 + nn] = wval;
    if (r < 8) oc2[r * 16 + nn] = wval * sS2[r];  // c2r = S2[:8,None]*V2[:8]
    sU1[nn * 16 + r] = wval;                      // W' for U2 compute
  }
  __syncthreads();

  // ---- phase 8: U2 = rest * W / S2 ;  c1r = U2[:, :8]
  float* oU2 = out + (size_t)B * 5648 + b * 2048;
  float* oc1 = out + (size_t)B * 128 + b * 1024;
  for (int o = tid; o < 2048; o += 256) {
    int row = o >> 4, r = o & 15;
    float acc = 0.0f;
#pragma unroll
    for (int nn = 0; nn < 16; ++nn)
      acc += sRest[row * 16 + nn] * sU1[nn * 16 + r];
    float sv = sS2[r];
    float rin = (sv > 1e-12f) ? (1.0f / sv) : 0.0f;
    float u2 = acc * rin;
    oU2[row * 16 + r] = u2;
    if (r < 8) oc1[row * 8 + r] = u2;
  }
}

extern "C" void kernel_launch(void* const* d_in, const int* in_sizes, int n_in,
                              void* d_out, int out_size, void* d_ws,
                              size_t ws_size, hipStream_t stream) {
  const float* x0 = (const float*)d_in[0];
  const float* x1 = (const float*)d_in[1];
  const float* x2 = (const float*)d_in[2];
  const float* w0 = (const float*)d_in[3];
  const float* w1 = (const float*)d_in[4];
  const float* w2 = (const float*)d_in[5];
  const float* b0 = (const float*)d_in[6];
  const float* b1 = (const float*)d_in[7];
  const float* b2 = (const float*)d_in[8];
  float* out = (float*)d_out;
  float* bfull = (float*)d_ws;  // 4096 floats
  int B = in_sizes[0] / 128;    // x0 is [B,1,16,8]

  bias_full_kernel<<<1, 256, 0, stream>>>(b0, b1, b2, bfull);
  ftt_solver_kernel<<<B, 256, 0, stream>>>(x0, x1, x2, w0, w1, w2, bfull, out, B);
}